// GAT_84799834292739
// MI455X (gfx1250) — compile-verified
//
#include <hip/hip_runtime.h>

#define NN 50000
#define EE 640000
#define HIDC 128
#define NH 8
#define CC 16
#define LLAYERS 3

typedef __attribute__((ext_vector_type(2))) float v2f;
typedef __attribute__((ext_vector_type(8))) float v8f;

// ---- monotonic float <-> ordered-uint mapping (for atomic max of floats) ----
__device__ __forceinline__ unsigned f2ord(float f) {
    unsigned u = __float_as_uint(f);
    return (u & 0x80000000u) ? ~u : (u | 0x80000000u);
}
__device__ __forceinline__ float ord2f(unsigned o) {
    unsigned u = (o & 0x80000000u) ? (o ^ 0x80000000u) : ~o;
    return __uint_as_float(u);
}

// ---- generic 32-bit fill ----
__global__ void fill_u32_kernel(unsigned* __restrict__ p, unsigned v, long long n) {
    long long i = (long long)blockIdx.x * blockDim.x + threadIdx.x;
    if (i < n) p[i] = v;
}

// ---- WMMA GEMM: Out[M,128] = A[M,K] @ W[K,128] (+ bias[128]) --------------
// grid.x = ceil((M/16)/4); block = 256 (8 waves). Each block covers 64 rows.
// Wave w owns cols [16w,16w+16) and computes FOUR 16x16 tiles, reusing each
// B fragment (loaded once per K-step) across 4 V_WMMA_F32_16X16X4_F32 ops.
template <int K>
__global__ void __launch_bounds__(256) gemm_wmma_kernel(
    const float* __restrict__ A, const float* __restrict__ W,
    const float* __restrict__ bias, float* __restrict__ Out, int M)
{
    __shared__ float sA[64][132];  // 132 % 64 = 4 -> conflict-free column reads; 132 % 4 = 0 -> float4 stores OK

    const int block_m = blockIdx.x * 64;
    const int tid = threadIdx.x;
    const int total_tiles = M >> 4;
    int nt = total_tiles - blockIdx.x * 4;   // valid 16-row tiles in this block
    if (nt > 4) nt = 4;

    // prefetch the shared weight panel (lowers to global_prefetch_b8)
    {
        int i = tid * 64;                     // one 256B line per thread
        if (i < K * HIDC) __builtin_prefetch(W + i, 0, 3);
    }

    // cooperatively stage the 64 x K A-panel into LDS with float4 loads;
    // clamp rows so the tail block never reads OOB (garbage rows never stored)
    constexpr int K4 = K / 4;
    for (int i = tid; i < 64 * K4; i += 256) {
        int r = i / K4, c4 = (i - r * K4) * 4;
        int row = block_m + r;
        if (row > M - 1) row = M - 1;
        float4 v = *(const float4*)(A + (size_t)row * K + c4);
        *(float4*)(&sA[r][c4]) = v;
    }
    __syncthreads();

    const int wave  = tid >> 5;
    const int lane  = tid & 31;
    const int n     = wave * 16 + (lane & 15);   // output column (B/C/D fragments)
    const int khalf = (lane >> 4) * 2;           // lanes 16-31 hold K+2,K+3
    const int mrow  = lane & 15;                 // A fragment row within a tile

    v8f acc[4] = {{}, {}, {}, {}};
    #pragma unroll
    for (int kk = 0; kk < K; kk += 4) {
        v2f b;
        b.x = W[(size_t)(kk + khalf) * HIDC + n];
        b.y = W[(size_t)(kk + khalf + 1) * HIDC + n];
        #pragma unroll
        for (int t = 0; t < 4; ++t) {
            v2f a;
            a.x = sA[t * 16 + mrow][kk + khalf];
            a.y = sA[t * 16 + mrow][kk + khalf + 1];
            acc[t] = __builtin_amdgcn_wmma_f32_16x16x4_f32(
                /*neg_a=*/false, a, /*neg_b=*/false, b,
                /*c_mod=*/(short)0, acc[t], /*reuse_a=*/false, /*reuse_b=*/false);
        }
    }

    const float bn = bias ? bias[n] : 0.0f;
    #pragma unroll
    for (int t = 0; t < 4; ++t) {
        if (t < nt) {   // wave-uniform guard: EXEC untouched, tail tiles skipped
            const int rbase = block_m + t * 16 + ((lane >> 4) << 3);
            #pragma unroll
            for (int r = 0; r < 8; ++r)
                Out[(size_t)(rbase + r) * HIDC + n] = acc[t][r] + bn;
        }
    }
}

// ---- per-(edge,head) attention logit + segment max over dst ---------------
__global__ void logits_max_kernel(
    const float* __restrict__ xl, const float* __restrict__ xr,
    const float* __restrict__ em, const int* __restrict__ src,
    const int* __restrict__ dst, const float* __restrict__ att,
    float* __restrict__ logits, unsigned* __restrict__ lmax)
{
    int t = blockIdx.x * blockDim.x + threadIdx.x;
    if (t >= EE * NH) return;
    int e = t >> 3, h = t & 7;
    int s = src[e], d = dst[e];
    const float4* pl = (const float4*)(xl + (size_t)s * HIDC + h * CC);
    const float4* pr = (const float4*)(xr + (size_t)d * HIDC + h * CC);
    const float4* pe = (const float4*)(em + (size_t)e * HIDC + h * CC);
    const float4* pa = (const float4*)(att + h * CC);
    float logit = 0.f;
    #pragma unroll
    for (int q = 0; q < 4; ++q) {
        float4 a4 = pl[q], b4 = pr[q], c4 = pe[q], w4 = pa[q];
        float vx = a4.x + b4.x + c4.x; vx = vx > 0.f ? vx : 0.2f * vx; logit += vx * w4.x;
        float vy = a4.y + b4.y + c4.y; vy = vy > 0.f ? vy : 0.2f * vy; logit += vy * w4.y;
        float vz = a4.z + b4.z + c4.z; vz = vz > 0.f ? vz : 0.2f * vz; logit += vz * w4.z;
        float vw = a4.w + b4.w + c4.w; vw = vw > 0.f ? vw : 0.2f * vw; logit += vw * w4.w;
    }
    logits[t] = logit;
    atomicMax(lmax + (size_t)d * NH + h, f2ord(logit));
}

// ---- p = exp(logit - lmax[dst]); denom[dst] += p (p overwrites logits) ----
__global__ void softmax_p_kernel(
    float* __restrict__ logits, const unsigned* __restrict__ lmax,
    float* __restrict__ denom, const int* __restrict__ dst)
{
    int t = blockIdx.x * blockDim.x + threadIdx.x;
    if (t >= EE * NH) return;
    int e = t >> 3, h = t & 7;
    int d = dst[e];
    float lm = ord2f(lmax[(size_t)d * NH + h]);
    float p = __expf(logits[t] - lm);
    logits[t] = p;
    atomicAdd(denom + (size_t)d * NH + h, p);
}

// ---- agg[dst] += xl[src] * alpha ------------------------------------------
__global__ void scatter_kernel(
    const float* __restrict__ p, const float* __restrict__ denom,
    const float* __restrict__ xl, const int* __restrict__ src,
    const int* __restrict__ dst, float* __restrict__ agg)
{
    int t = blockIdx.x * blockDim.x + threadIdx.x;
    if (t >= EE * NH) return;
    int e = t >> 3, h = t & 7;
    int s = src[e], d = dst[e];
    float alpha = p[t] / (denom[(size_t)d * NH + h] + 1e-16f);
    const float4* px = (const float4*)(xl + (size_t)s * HIDC + h * CC);
    float* pd = agg + (size_t)d * HIDC + h * CC;
    #pragma unroll
    for (int q = 0; q < 4; ++q) {
        float4 v = px[q];
        atomicAdd(pd + q * 4 + 0, v.x * alpha);
        atomicAdd(pd + q * 4 + 1, v.y * alpha);
        atomicAdd(pd + q * 4 + 2, v.z * alpha);
        atomicAdd(pd + q * 4 + 3, v.w * alpha);
    }
}

// ---- LayerNorm(agg + bias_o) -> ReLU -> + residual; one wave32 per node ---
__global__ void __launch_bounds__(256) ln_res_kernel(
    const float* __restrict__ agg, const float* __restrict__ bias_o,
    const float* __restrict__ g, const float* __restrict__ b,
    const float* __restrict__ xin, float* __restrict__ xout)
{
    int node = blockIdx.x * 8 + (threadIdx.x >> 5);
    int lane = threadIdx.x & 31;
    if (node >= NN) return;
    int c0 = lane * 4;

    float4 v  = *(const float4*)(agg + (size_t)node * HIDC + c0);
    float4 bo = *(const float4*)(bias_o + c0);
    v.x += bo.x; v.y += bo.y; v.z += bo.z; v.w += bo.w;

    float s  = v.x + v.y + v.z + v.w;
    float sq = v.x * v.x + v.y * v.y + v.z * v.z + v.w * v.w;
    #pragma unroll
    for (int m = 16; m >= 1; m >>= 1) {
        s  += __shfl_xor(s,  m, 32);
        sq += __shfl_xor(sq, m, 32);
    }
    float mu  = s * (1.0f / 128.0f);
    float var = sq * (1.0f / 128.0f) - mu * mu;
    float inv = rsqrtf(var + 1e-5f);

    float4 gg = *(const float4*)(g + c0);
    float4 bb = *(const float4*)(b + c0);
    float4 xi = *(const float4*)(xin + (size_t)node * HIDC + c0);
    float4 o;
    o.x = fmaxf(gg.x * (v.x - mu) * inv + bb.x, 0.f) + xi.x;
    o.y = fmaxf(gg.y * (v.y - mu) * inv + bb.y, 0.f) + xi.y;
    o.z = fmaxf(gg.z * (v.z - mu) * inv + bb.z, 0.f) + xi.z;
    o.w = fmaxf(gg.w * (v.w - mu) * inv + bb.w, 0.f) + xi.w;
    *(float4*)(xout + (size_t)node * HIDC + c0) = o;
}

extern "C" void kernel_launch(void* const* d_in, const int* in_sizes, int n_in,
                              void* d_out, int out_size, void* d_ws, size_t ws_size,
                              hipStream_t stream) {
    const float* x0        = (const float*)d_in[0];
    const int*   eidx      = (const int*)d_in[2];
    const int*   src       = eidx;
    const int*   dst       = eidx + EE;
    const float* edge_attr = (const float*)d_in[3];
    const float* Wt        = (const float*)d_in[4];
    const float* bt        = (const float*)d_in[5];
    const float* Wl        = (const float*)d_in[6];
    const float* bl        = (const float*)d_in[7];
    const float* Wr        = (const float*)d_in[8];
    const float* br        = (const float*)d_in[9];
    const float* We        = (const float*)d_in[10];
    const float* att       = (const float*)d_in[11];
    const float* bias_o    = (const float*)d_in[12];
    const float* ln_g      = (const float*)d_in[13];
    const float* ln_b      = (const float*)d_in[14];

    float* ws = (float*)d_ws;
    size_t off = 0;
    float*    ea    = ws + off; off += (size_t)EE * HIDC;
    float*    em    = ws + off; off += (size_t)EE * HIDC;
    float*    xl    = ws + off; off += (size_t)NN * HIDC;
    float*    xr    = ws + off; off += (size_t)NN * HIDC;
    float*    lg    = ws + off; off += (size_t)EE * NH;
    unsigned* lmax  = (unsigned*)(ws + off); off += (size_t)NN * NH;
    float*    denom = ws + off; off += (size_t)NN * NH;
    float*    agg   = ws + off; off += (size_t)NN * HIDC;
    float*    xbuf  = ws + off; off += (size_t)NN * HIDC;

    dim3 blk(256);
    const int eth   = EE * NH;                 // 5,120,000
    const int egrid = eth / 256;               // 20000 exact
    const int ngemm = ((NN / 16) + 3) / 4;     // 782 blocks (64 rows each, tail clamped)
    const int egemm = (EE / 16) / 4;           // 10000 exact

    // shared edge transform: ea = edge_attr @ Wt + bt
    gemm_wmma_kernel<16><<<egemm, blk, 0, stream>>>(edge_attr, Wt, bt, ea, EE);

    const float* xcur = x0;
    for (int l = 0; l < LLAYERS; ++l) {
        const float* WlL = Wl + (size_t)l * HIDC * HIDC;
        const float* WrL = Wr + (size_t)l * HIDC * HIDC;
        const float* WeL = We + (size_t)l * HIDC * HIDC;

        gemm_wmma_kernel<128><<<ngemm, blk, 0, stream>>>(xcur, WlL, bl + (size_t)l * HIDC, xl, NN);
        gemm_wmma_kernel<128><<<ngemm, blk, 0, stream>>>(xcur, WrL, br + (size_t)l * HIDC, xr, NN);
        gemm_wmma_kernel<128><<<egemm, blk, 0, stream>>>(ea, WeL, nullptr, em, EE);

        fill_u32_kernel<<<((NN * NH) + 255) / 256, blk, 0, stream>>>(lmax, 0u, (long long)NN * NH);
        fill_u32_kernel<<<((NN * NH) + 255) / 256, blk, 0, stream>>>((unsigned*)denom, 0u, (long long)NN * NH);
        fill_u32_kernel<<<((NN * HIDC) + 255) / 256, blk, 0, stream>>>((unsigned*)agg, 0u, (long long)NN * HIDC);

        logits_max_kernel<<<egrid, blk, 0, stream>>>(xl, xr, em, src, dst,
                                                     att + (size_t)l * NH * CC, lg, lmax);
        softmax_p_kernel<<<egrid, blk, 0, stream>>>(lg, lmax, denom, dst);
        scatter_kernel<<<egrid, blk, 0, stream>>>(lg, denom, xl, src, dst, agg);

        float* xout = (l == LLAYERS - 1) ? (float*)d_out : xbuf;
        ln_res_kernel<<<NN / 8, blk, 0, stream>>>(agg, bias_o + (size_t)l * HIDC,
                                                  ln_g + (size_t)l * HIDC, ln_b + (size_t)l * HIDC,
                                                  xcur, xout);
        xcur = xout;
    }
}